// Nlplant_76510547411365
// MI455X (gfx1250) — compile-verified
//
#include <hip/hip_runtime.h>
#include <math.h>

#define THREADS 256
#define L_TAB3 0
#define L_TAB2 9500
#define L_TAB1 15580
#define L_ETA  16000
#define L_TOT  16008   // 64 KB LDS: full table copy per workgroup

// ---- fast hardware transcendentals (v_sin/v_cos take revolutions) ----
#if defined(__AMDGCN__) && __has_builtin(__builtin_amdgcn_sinf)
#define FSIN(x)  __builtin_amdgcn_sinf((x) * 0.15915494309189535f)
#define FCOS(x)  __builtin_amdgcn_cosf((x) * 0.15915494309189535f)
#define FRCP(x)  __builtin_amdgcn_rcpf(x)
#define FPOW(x, y) __builtin_amdgcn_exp2f((y) * __builtin_amdgcn_logf(x))  // x > 0
#else
#define FSIN(x)  sinf(x)
#define FCOS(x)  cosf(x)
#define FRCP(x)  (1.0f / (x))
#define FPOW(x, y) powf((x), (y))
#endif

__device__ __forceinline__ void grid_idx(float x, float lo, float scale, int nm1,
                                         int& i0, float& f) {
    // t = clip((x-lo)/(hi-lo)*(n-1), 0, n-1); i0 = min(int(t), n-2); f = t - i0
    float t = (x - lo) * scale;
    t = fminf(fmaxf(t, 0.0f), (float)nm1);
    int i = (int)t;
    i = (i > nm1 - 1) ? (nm1 - 1) : i;
    i0 = i;
    f = t - (float)i;
}

__global__ __launch_bounds__(THREADS) void nlplant_kernel(
    const float* __restrict__ xu, const float* __restrict__ tab3,
    const float* __restrict__ tab2, const float* __restrict__ tab1,
    const float* __restrict__ tab_eta, float* __restrict__ out, int B)
{
    __shared__ __align__(16) float lut[L_TOT];
    const int tid = threadIdx.x;

    // ---- Stage all tables into LDS (CDNA5 async-to-LDS path when available) ----
#if defined(__gfx1250__) && __has_builtin(__builtin_amdgcn_global_load_async_to_lds_b128)
    typedef int v4i __attribute__((ext_vector_type(4)));
    typedef __attribute__((address_space(1))) v4i gv4i;
    typedef __attribute__((address_space(3))) v4i sv4i;
    // generic->AS1 is value-identical; generic LDS ptr low 32 bits == LDS offset (AS3)
    #define CP16(gsrc, ldst)                                                     \
        __builtin_amdgcn_global_load_async_to_lds_b128(                          \
            (gv4i*)(unsigned long long)(gsrc),                                   \
            (sv4i*)(unsigned)(unsigned long long)(ldst), 0, 0)
    for (int i = tid; i < 2375; i += THREADS) CP16(tab3 + 4 * i, &lut[L_TAB3 + 4 * i]);
    for (int i = tid; i < 1520; i += THREADS) CP16(tab2 + 4 * i, &lut[L_TAB2 + 4 * i]);
    for (int i = tid; i < 105;  i += THREADS) CP16(tab1 + 4 * i, &lut[L_TAB1 + 4 * i]);
    if (tid == 0) CP16(tab_eta, &lut[L_ETA]);
    if (tid == 1) lut[L_ETA + 4] = tab_eta[4];
    #if __has_builtin(__builtin_amdgcn_s_wait_asynccnt)
    __builtin_amdgcn_s_wait_asynccnt(0);
    #else
    asm volatile("s_wait_asynccnt 0" ::: "memory");
    #endif
    #undef CP16
#else
    for (int i = tid; i < 9500; i += THREADS) lut[L_TAB3 + i] = tab3[i];
    for (int i = tid; i < 6080; i += THREADS) lut[L_TAB2 + i] = tab2[i];
    for (int i = tid; i < 420;  i += THREADS) lut[L_TAB1 + i] = tab1[i];
    if (tid < 5) lut[L_ETA + tid] = tab_eta[tid];
#endif
    __syncthreads();

    const float r2d = 57.29577951308232f;
    const float g = 32.17f, Bspan = 30.0f, S = 300.0f, cbar = 11.32f;
    const float xcgr = 0.35f, xcg = 0.25f;
    const float Jy = 55814.0f, Jxz = 982.0f, Jz = 63100.0f, Jx = 9496.0f;
    const float inv_m  = 1.0f / 636.94f;
    const float inv_Jy = 1.0f / Jy;
    const float inv_denom = 1.0f / (Jx * Jz - Jxz * Jxz);

    for (int row = blockIdx.x * THREADS + tid; row < B; row += gridDim.x * THREADS) {
        const float* x = xu + (size_t)row * 17;
        float alt = x[2], phi = x[3], theta = x[4], psi = x[5];
        float vt = fmaxf(x[6], 0.01f);
        float alpha = x[7], beta = x[8];
        float P = x[9], Q = x[10], R = x[11];
        float T = x[12], el = x[13], ail = x[14], rud = x[15], lef = x[16];

        float a_deg = alpha * r2d, b_deg = beta * r2d;
        float dail = ail * (1.0f / 21.5f);
        float drud = rud * (1.0f / 30.0f);
        float dlef = 1.0f - lef * (1.0f / 25.0f);

        float tfac = 1.0f - 7.03e-6f * alt;
        float rho = 0.002377f * FPOW(tfac, 4.14f);
        float qbar = 0.5f * rho * vt * vt;

        float sa = FSIN(alpha), ca = FCOS(alpha);
        float sb = FSIN(beta),  cb = FCOS(beta);
        float st = FSIN(theta), ct = FCOS(theta);
        float sphi = FSIN(phi), cphi = FCOS(phi);
        float spsi = FSIN(psi), cpsi = FCOS(psi);

        float rcp_ct = FRCP(ct);
        float rcp_vt = FRCP(vt);
        float rcp_cb = FRCP(cb);
        float tt = st * rcp_ct;

        float U = vt * ca * cb, V = vt * sb, W = vt * sa * cb;
        float xd0 = U * (ct * cpsi) + V * (sphi * cpsi * st - cphi * spsi) + W * (cphi * st * cpsi + sphi * spsi);
        float xd1 = U * (ct * spsi) + V * (sphi * spsi * st + cphi * cpsi) + W * (cphi * st * spsi - sphi * cpsi);
        float xd2 = U * st - V * (sphi * ct) - W * (cphi * ct);
        float xd3 = P + tt * (Q * sphi + R * cphi);
        float xd4 = Q * cphi - R * sphi;
        float xd5 = (Q * sphi + R * cphi) * rcp_ct;

        // ---- grid indices / weights ----
        int ia, ib, ie; float fa, fb, fe;
        grid_idx(a_deg, -20.0f, 19.0f / 110.0f, 19, ia, fa);   // n=20
        grid_idx(b_deg, -30.0f, 18.0f / 60.0f,  18, ib, fb);   // n=19
        grid_idx(el,    -25.0f, 4.0f / 50.0f,    4, ie, fe);   // n=5
        float ga = 1.0f - fa, gb = 1.0f - fb, ge = 1.0f - fe;
        float w00 = ga * gb, w01 = ga * fb, w10 = fa * gb, w11 = fa * fb;

        // ---- LDS gathers: tab3 trilinear (+ el=0 slice at node k=2), tab2 bilinear, tab1 linear ----
        float v3[5], v30[5];
        int base_ab3 = ia * 95 + ib * 5;
        #pragma unroll
        for (int c = 0; c < 5; ++c) {
            const float* p = lut + L_TAB3 + c * 1900 + base_ab3;
            float t00 = p[ie] * ge       + p[ie + 1] * fe;
            float t01 = p[5 + ie] * ge   + p[6 + ie] * fe;
            float t10 = p[95 + ie] * ge  + p[96 + ie] * fe;
            float t11 = p[100 + ie] * ge + p[101 + ie] * fe;
            v3[c]  = w00 * t00 + w01 * t01 + w10 * t10 + w11 * t11;
            v30[c] = w00 * p[2] + w01 * p[7] + w10 * p[97] + w11 * p[102];
        }
        float v2[16];
        int base_ab2 = ia * 19 + ib;
        #pragma unroll
        for (int c = 0; c < 16; ++c) {
            const float* p = lut + L_TAB2 + c * 380 + base_ab2;
            v2[c] = w00 * p[0] + w01 * p[1] + w10 * p[19] + w11 * p[20];
        }
        float v1[21];
        #pragma unroll
        for (int c = 0; c < 21; ++c) {
            const float* p = lut + L_TAB1 + c * 20 + ia;
            v1[c] = p[0] * ga + p[1] * fa;
        }
        float eta_el = lut[L_ETA + ie] * ge + lut[L_ETA + ie + 1] * fe;

        // ---- aero coefficient build-up ----
        float Cx = v3[0], Cz = v3[1], Cm = v3[2], Cn = v3[3], Cl = v3[4];
        float Cy = v2[0];
        float d_Cx_lef = v2[1] - v30[0];
        float d_Cz_lef = v2[2] - v30[1];
        float d_Cm_lef = v2[3] - v30[2];
        float d_Cy_lef = v2[4] - v2[0];
        float d_Cn_lef = v2[5] - v30[3];
        float d_Cl_lef = v2[6] - v30[4];
        float d_Cy_r30 = v2[7] - v2[0];
        float d_Cn_r30 = v2[8] - v30[3];
        float d_Cl_r30 = v2[9] - v30[4];
        float d_Cy_a20     = v2[10] - v2[0];
        float d_Cy_a20_lef = v2[11] - v2[4];
        float d_Cn_a20     = v2[12] - v30[3];
        float d_Cn_a20_lef = v2[13] - v2[5];
        float d_Cl_a20     = v2[14] - v30[4];
        float d_Cl_a20_lef = v2[15] - v2[6];

        float c2v = 0.5f * rcp_vt;
        float dXdQ = cbar * c2v * (v1[0] + v1[9] * dlef);
        float Cx_tot = Cx + d_Cx_lef * dlef + dXdQ * Q;
        float dZdQ = cbar * c2v * (v1[3] + v1[12] * dlef);
        float Cz_tot = Cz + d_Cz_lef * dlef + dZdQ * Q;
        float dMdQ = cbar * c2v * (v1[6] + v1[15] * dlef);
        float Cm_tot = Cm * eta_el + Cz_tot * (xcgr - xcg) + d_Cm_lef * dlef + dMdQ * Q + v1[20];
        float dYdail = d_Cy_a20 + d_Cy_a20_lef * dlef;
        float dYdR = Bspan * c2v * (v1[1] + v1[10] * dlef);
        float dYdP = Bspan * c2v * (v1[2] + v1[11] * dlef);
        float Cy_tot = Cy + d_Cy_lef * dlef + dYdail * dail + d_Cy_r30 * drud + dYdR * R + dYdP * P;
        float dNdail = d_Cn_a20 + d_Cn_a20_lef * dlef;
        float dNdR = Bspan * c2v * (v1[7] + v1[16] * dlef);
        float dNdP = Bspan * c2v * (v1[8] + v1[17] * dlef);
        float Cn_tot = Cn + d_Cn_lef * dlef - Cy_tot * (xcgr - xcg) * (cbar / Bspan)
                     + dNdail * dail + d_Cn_r30 * drud + dNdR * R + dNdP * P + v1[18] * b_deg;
        float dLdail = d_Cl_a20 + d_Cl_a20_lef * dlef;
        float dLdR = Bspan * c2v * (v1[4] + v1[13] * dlef);
        float dLdP = Bspan * c2v * (v1[5] + v1[14] * dlef);
        float Cl_tot = Cl + d_Cl_lef * dlef + dLdail * dail + d_Cl_r30 * drud
                     + dLdR * R + dLdP * P + v1[19] * b_deg;

        // ---- equations of motion ----
        float qS_m = qbar * S * inv_m;
        float Udot = R * V - Q * W - g * st + qS_m * Cx_tot + T * inv_m;
        float Vdot = P * W - R * U + g * ct * sphi + qS_m * Cy_tot;
        float Wdot = Q * U - P * V + g * ct * cphi + qS_m * Cz_tot;
        float vt_dot = (U * Udot + V * Vdot + W * Wdot) * rcp_vt;
        float alpha_dot = (U * Wdot - W * Udot) * FRCP(U * U + W * W);
        float beta_dot = (Vdot * vt - V * vt_dot) * (rcp_vt * rcp_vt) * rcp_cb;
        float L_tot = Cl_tot * qbar * S * Bspan;
        float M_tot = Cm_tot * qbar * S * cbar;
        float N_tot = Cn_tot * qbar * S * Bspan;
        float P_dot = (Jz * L_tot + Jxz * N_tot - (Jz * (Jz - Jy) + Jxz * Jxz) * Q * R
                      + Jxz * (Jx - Jy + Jz) * P * Q) * inv_denom;
        float Q_dot = (M_tot + (Jz - Jx) * P * R - Jxz * (P * P - R * R)) * inv_Jy;
        float R_dot = (Jx * N_tot + Jxz * L_tot + (Jx * (Jx - Jy) + Jxz * Jxz) * P * Q
                      - Jxz * (Jx - Jy + Jz) * Q * R) * inv_denom;

        float* o = out + (size_t)row * 17;
        o[0] = xd0;  o[1] = xd1;  o[2] = xd2;  o[3] = xd3;
        o[4] = xd4;  o[5] = xd5;  o[6] = vt_dot; o[7] = alpha_dot;
        o[8] = beta_dot; o[9] = P_dot; o[10] = Q_dot; o[11] = R_dot;
        o[12] = 0.0f; o[13] = 0.0f; o[14] = 0.0f; o[15] = 0.0f; o[16] = 0.0f;
    }
}

extern "C" void kernel_launch(void* const* d_in, const int* in_sizes, int n_in,
                              void* d_out, int out_size, void* d_ws, size_t ws_size,
                              hipStream_t stream) {
    (void)n_in; (void)out_size; (void)d_ws; (void)ws_size;
    const float* xu      = (const float*)d_in[0];
    const float* tab3    = (const float*)d_in[1];
    const float* tab2    = (const float*)d_in[2];
    const float* tab1    = (const float*)d_in[3];
    const float* tab_eta = (const float*)d_in[4];
    float* out = (float*)d_out;

    int B = in_sizes[0] / 17;
    int blocks = (B + THREADS - 1) / THREADS;
    if (blocks > 1024) blocks = 1024;   // ~4 blocks/WGP by LDS; grid-stride covers the rest
    if (blocks < 1) blocks = 1;
    nlplant_kernel<<<blocks, THREADS, 0, stream>>>(xu, tab3, tab2, tab1, tab_eta, out, B);
}